// Pool_90374701842962
// MI455X (gfx1250) — compile-verified
//
#include <hip/hip_runtime.h>
#include <hip/hip_bf16.h>

// ---------------------------------------------------------------------------
// TopK graph pooling (K=0.5). Pipeline:
//   1) scores = sigmoid(x @ W + b)  via V_WMMA_F32_16X16X4_F32 (exact f32)
//   2) exact 50000th-order-statistic threshold via 2-pass 16-bit radix select
//   3) stable node compaction (block scans) -> pool_indices / mapping
//   4) gather new_x, new_pos
//   5) stable edge partition (kept-first), remap endpoints, permute edge_attr
// ---------------------------------------------------------------------------

typedef __attribute__((ext_vector_type(2))) float v2f;
typedef __attribute__((ext_vector_type(8))) float v8f;

#if defined(__has_builtin)
#if __has_builtin(__builtin_amdgcn_wmma_f32_16x16x4_f32)
#define HAVE_WMMA_F32X4 1
#endif
#endif

#define CTRL_BUCKET   0
#define CTRL_CUMBEF   1
#define CTRL_THRKEY   2
#define CTRL_NKEPT_E  3

__device__ __forceinline__ unsigned f2key(float s) {
    unsigned u = __float_as_uint(s);
    return (u & 0x80000000u) ? ~u : (u | 0x80000000u);
}

// ---------------- scores: GEMV (N x 128) @ (128 x 1) via WMMA -------------
__global__ void k_scores(const float* __restrict__ x, const float* __restrict__ W,
                         const float* __restrict__ bias, unsigned* __restrict__ keys,
                         int N) {
    int wave = threadIdx.x >> 5;                      // 8 waves / block
    int lane = threadIdx.x & 31;
    long long rowBase = ((long long)blockIdx.x * 8 + wave) * 16;
    if (rowBase >= N) return;                         // wave-uniform exit
    float bv = bias[0];
#if HAVE_WMMA_F32X4
    int m  = lane & 15;
    int kp = (lane >> 4) * 2;                         // 0 or 2
    long long row = rowBase + m;
    if (row >= N) row = N - 1;                        // clamp (keep EXEC full)
    const float* xr = x + row * 128;
    v8f c = {};
    for (int k0 = 0; k0 < 128; k0 += 4) {
        v2f a, b;
        a.x = xr[k0 + kp];
        a.y = xr[k0 + kp + 1];
        b.x = W[k0 + kp];                             // broadcast W to all 16 cols
        b.y = W[k0 + kp + 1];
        c = __builtin_amdgcn_wmma_f32_16x16x4_f32(false, a, false, b,
                                                  (short)0, c, false, false);
    }
    // C layout: lanes 0-15 -> M=v (VGPR v), N=lane; lanes 16-31 -> M=v+8.
    // Column N=0 lives in lanes 0 and 16.
    if ((lane & 15) == 0) {
        int mbase = (lane >> 4) * 8;
        for (int v = 0; v < 8; ++v) {
            long long r = rowBase + mbase + v;
            if (r < N) {
                float s = 1.0f / (1.0f + __expf(-(c[v] + bv)));
                keys[r] = f2key(s);
            }
        }
    }
#else
    // Fallback: one lane per row, 16 rows per wave handled by lanes 0..15
    if (lane < 16) {
        long long r = rowBase + lane;
        if (r < N) {
            const float* xr = x + r * 128;
            float acc = 0.f;
            for (int k = 0; k < 128; ++k) acc = fmaf(xr[k], W[k], acc);
            float s = 1.0f / (1.0f + __expf(-(acc + bv)));
            keys[r] = f2key(s);
        }
    }
#endif
}

// ---------------- utility: zero ----------------
__global__ void k_zero_u32(unsigned* p, long long n) {
    long long i = (long long)blockIdx.x * blockDim.x + threadIdx.x;
    if (i < n) p[i] = 0;
}

// ---------------- radix select ----------------
__global__ void k_hist1(const unsigned* __restrict__ keys, int N, unsigned* hist) {
    long long i = (long long)blockIdx.x * blockDim.x + threadIdx.x;
    if (i < N) atomicAdd(&hist[keys[i] >> 16], 1u);
}

__global__ void k_hist2(const unsigned* __restrict__ keys, int N,
                        const unsigned* __restrict__ ctrl, unsigned* hist) {
    long long i = (long long)blockIdx.x * blockDim.x + threadIdx.x;
    if (i < N) {
        unsigned k = keys[i];
        if ((k >> 16) == ctrl[CTRL_BUCKET]) atomicAdd(&hist[k & 0xFFFFu], 1u);
    }
}

// single block, 1024 threads; finds bin containing rank `target` (1-indexed)
__global__ void k_find(const unsigned* __restrict__ hist, unsigned R, int mode,
                       unsigned* ctrl) {
    __shared__ unsigned s[1024];
    __shared__ unsigned base;
    int t = threadIdx.x;
    if (t == 0) base = 0;
    __syncthreads();
    unsigned target = (mode == 0) ? R : (R - ctrl[CTRL_CUMBEF]);
    for (int start = 0; start < 65536; start += 1024) {
        int i = start + t;
        unsigned v = hist[i];
        s[t] = v; __syncthreads();
        for (int o = 1; o < 1024; o <<= 1) {
            unsigned xx = (t >= o) ? s[t - o] : 0u;
            __syncthreads();
            s[t] += xx;
            __syncthreads();
        }
        unsigned cumExcl = base + s[t] - v;
        if (cumExcl < target && target <= cumExcl + v) {
            if (mode == 0) { ctrl[CTRL_BUCKET] = (unsigned)i; ctrl[CTRL_CUMBEF] = cumExcl; }
            else           { ctrl[CTRL_THRKEY] = (ctrl[CTRL_BUCKET] << 16) | (unsigned)i; }
        }
        __syncthreads();
        if (t == 0) base += s[1023];
        __syncthreads();
    }
}

// ---------------- block scan helper (256 threads) ----------------
__device__ __forceinline__ unsigned blockExclScan256(unsigned v, unsigned* s,
                                                     unsigned* totalOut) {
    int t = threadIdx.x;
    s[t] = v; __syncthreads();
    for (int o = 1; o < 256; o <<= 1) {
        unsigned xx = (t >= o) ? s[t - o] : 0u;
        __syncthreads();
        s[t] += xx;
        __syncthreads();
    }
    if (totalOut) *totalOut = s[255];
    return s[t] - v;
}

// ---------------- generic single-block scan of M block counts ----------------
__global__ void k_scan_blocks(const unsigned* __restrict__ cnt, unsigned* off, int M,
                              unsigned* totalOut, float* totalOutF) {
    __shared__ unsigned s[1024];
    __shared__ unsigned base;
    int t = threadIdx.x;
    if (t == 0) base = 0;
    __syncthreads();
    for (int start = 0; start < M; start += 1024) {
        int i = start + t;
        unsigned v = (i < M) ? cnt[i] : 0u;
        s[t] = v; __syncthreads();
        for (int o = 1; o < 1024; o <<= 1) {
            unsigned xx = (t >= o) ? s[t - o] : 0u;
            __syncthreads();
            s[t] += xx;
            __syncthreads();
        }
        if (i < M) off[i] = base + s[t] - v;
        __syncthreads();
        if (t == 0) base += s[1023];
        __syncthreads();
    }
    if (t == 0) {
        if (totalOut)  *totalOut  = base;
        if (totalOutF) *totalOutF = (float)base;
    }
}

// ---------------- node compaction ----------------
__global__ void k_node_count(const unsigned* __restrict__ keys, int N,
                             const unsigned* __restrict__ ctrl, unsigned* blkCnt) {
    __shared__ unsigned s[256];
    unsigned thr = ctrl[CTRL_THRKEY];
    long long base = (long long)blockIdx.x * 1024;
    unsigned loc = 0;
    for (int it = 0; it < 4; ++it) {
        long long i = base + (long long)threadIdx.x * 4 + it;
        if (i < N && keys[i] >= thr) loc++;
    }
    unsigned total;
    (void)blockExclScan256(loc, s, &total);
    if (threadIdx.x == 0) blkCnt[blockIdx.x] = total;
}

__global__ void k_node_scatter(const unsigned* __restrict__ keys, int N, int numKeep,
                               const unsigned* __restrict__ ctrl,
                               const unsigned* __restrict__ blkOff,
                               int* __restrict__ mapping, int* __restrict__ poolIdx,
                               float* __restrict__ outPool) {
    __shared__ unsigned s[256];
    unsigned thr = ctrl[CTRL_THRKEY];
    long long base = (long long)blockIdx.x * 1024;
    int f[4]; unsigned loc = 0;
    for (int it = 0; it < 4; ++it) {
        long long i = base + (long long)threadIdx.x * 4 + it;
        f[it] = (i < N && keys[i] >= thr) ? 1 : 0;
        loc += f[it];
    }
    unsigned excl = blockExclScan256(loc, s, nullptr);
    unsigned off = blkOff[blockIdx.x];
    unsigned run = excl;
    for (int it = 0; it < 4; ++it) {
        long long i = base + (long long)threadIdx.x * 4 + it;
        if (i < N) {
            unsigned rank = off + run;
            if (f[it] && rank < (unsigned)numKeep) {
                mapping[i] = (int)rank;
                poolIdx[rank] = (int)i;
                outPool[rank] = (float)i;
            } else {
                mapping[i] = -1;
            }
            run += f[it];
        }
    }
}

// ---------------- node gathers ----------------
__global__ void k_gather_x(const float* __restrict__ x, const int* __restrict__ poolIdx,
                           float* __restrict__ outX, int numKeep) {
    long long t = (long long)blockIdx.x * blockDim.x + threadIdx.x;
    long long total = (long long)numKeep * 32;        // float4 chunks
    if (t >= total) return;
    long long r = t >> 5; int q = (int)(t & 31);
    long long src = poolIdx[r];
    const float4* in  = (const float4*)(x    + src * 128 + q * 4);
    float4*       out = (float4*)(outX + r   * 128 + q * 4);
    *out = *in;
}

__global__ void k_gather_pos(const float* __restrict__ pos, const int* __restrict__ poolIdx,
                             float* __restrict__ outPos, int numKeep) {
    long long t = (long long)blockIdx.x * blockDim.x + threadIdx.x;
    long long total = (long long)numKeep * 3;
    if (t >= total) return;
    long long r = t / 3; int c = (int)(t % 3);
    outPos[t] = pos[(long long)poolIdx[r] * 3 + c];
}

// ---------------- edge partition ----------------
__global__ void k_edge_count(const int* __restrict__ ei, const int* __restrict__ mapping,
                             long long E, unsigned* blkCnt) {
    __shared__ unsigned s[256];
    long long base = (long long)blockIdx.x * 1024;
    unsigned loc = 0;
    for (int it = 0; it < 4; ++it) {
        long long e = base + (long long)threadIdx.x * 4 + it;
        if (e < E) {
            if (mapping[ei[e]] >= 0 && mapping[ei[E + e]] >= 0) loc++;
        }
    }
    unsigned total;
    (void)blockExclScan256(loc, s, &total);
    if (threadIdx.x == 0) blkCnt[blockIdx.x] = total;
}

__global__ void k_edge_scatter(const int* __restrict__ ei, const int* __restrict__ mapping,
                               long long E, const unsigned* __restrict__ blkOff,
                               const unsigned* __restrict__ ctrl,
                               unsigned* __restrict__ dstIdx,
                               float* __restrict__ outEI) {
    __shared__ unsigned s[256];
    long long base = (long long)blockIdx.x * 1024;
    unsigned numKept = ctrl[CTRL_NKEPT_E];
    int f[4]; unsigned loc = 0;
    for (int it = 0; it < 4; ++it) {
        long long e = base + (long long)threadIdx.x * 4 + it;
        f[it] = (e < E && mapping[ei[e]] >= 0 && mapping[ei[E + e]] >= 0) ? 1 : 0;
        loc += f[it];
    }
    unsigned excl = blockExclScan256(loc, s, nullptr);
    unsigned off = blkOff[blockIdx.x];
    unsigned run = excl;
    for (int it = 0; it < 4; ++it) {
        long long e = base + (long long)threadIdx.x * 4 + it;
        if (e < E) {
            unsigned keptBefore = off + run;                   // exclusive kept prefix
            unsigned j = f[it] ? keptBefore
                               : numKept + (unsigned)(e - (long long)keptBefore);
            dstIdx[e] = j;
            outEI[j]     = (float)mapping[ei[e]];
            outEI[E + j] = (float)mapping[ei[E + e]];
            run += f[it];
        }
    }
}

__global__ void k_attr_scatter(const float* __restrict__ attr,
                               const unsigned* __restrict__ dstIdx,
                               float* __restrict__ outAttr, long long E) {
    long long t = (long long)blockIdx.x * blockDim.x + threadIdx.x;
    if (t >= E * 4) return;
    long long e = t >> 2; int c = (int)(t & 3);
    long long j = dstIdx[e];
    const float4* in  = (const float4*)(attr    + e * 16 + c * 4);
    float4*       out = (float4*)(outAttr + j * 16 + c * 4);
    *out = *in;
}

// ---------------------------------------------------------------------------
extern "C" void kernel_launch(void* const* d_in, const int* in_sizes, int n_in,
                              void* d_out, int out_size, void* d_ws, size_t ws_size,
                              hipStream_t stream) {
    const float* x    = (const float*)d_in[0];
    const int*   ei   = (const int*)d_in[1];
    const float* attr = (const float*)d_in[2];
    const float* pos  = (const float*)d_in[3];
    const float* W    = (const float*)d_in[4];
    const float* bias = (const float*)d_in[5];

    const int       N  = in_sizes[0] / 128;
    const long long E  = (long long)in_sizes[1] / 2;
    const int       NK = N / 2;              // num_keep == kth rank (K = 0.5)
    const unsigned  R  = (unsigned)NK;       // 1-indexed rank of threshold

    // ---- workspace (bump allocator, 256B aligned) ----
    char* w = (char*)d_ws;
    auto alloc = [&](size_t bytes) -> void* {
        void* p = (void*)w;
        w += (bytes + 255) & ~(size_t)255;
        return p;
    };
    unsigned* keys    = (unsigned*)alloc((size_t)N * 4);
    int*      mapping = (int*)     alloc((size_t)N * 4);
    int*      poolIdx = (int*)     alloc((size_t)NK * 4);
    unsigned* hist1   = (unsigned*)alloc(65536 * 4);
    unsigned* hist2   = (unsigned*)alloc(65536 * 4);
    unsigned* ctrl    = (unsigned*)alloc(256);
    const int nbN = (N + 1023) / 1024;
    const int nbE = (int)((E + 1023) / 1024);
    unsigned* nBlkCnt = (unsigned*)alloc((size_t)nbN * 4);
    unsigned* nBlkOff = (unsigned*)alloc((size_t)nbN * 4);
    unsigned* eBlkCnt = (unsigned*)alloc((size_t)nbE * 4);
    unsigned* eBlkOff = (unsigned*)alloc((size_t)nbE * 4);
    unsigned* dstIdx  = (unsigned*)alloc((size_t)E * 4);

    // ---- output layout (concatenated f32) ----
    float* out      = (float*)d_out;
    float* outX     = out;                               // NK*128
    float* outEI    = outX  + (long long)NK * 128;       // 2*E
    float* outAttr  = outEI + 2 * E;                     // E*16
    float* outPos   = outAttr + E * 16;                  // NK*3
    float* outPool  = outPos + (long long)NK * 3;        // NK
    float* outCnt   = outPool + NK;                      // 1

    // 1) zero histograms (131072 u32)
    k_zero_u32<<<(131072 + 255) / 256, 256, 0, stream>>>(hist1, 131072);

    // 2) scores -> monotone keys (WMMA GEMV)
    k_scores<<<(N + 127) / 128, 256, 0, stream>>>(x, W, bias, keys, N);

    // 3) radix select: exact 50000th-smallest key
    k_hist1<<<(N + 255) / 256, 256, 0, stream>>>(keys, N, hist1);
    k_find<<<1, 1024, 0, stream>>>(hist1, R, 0, ctrl);
    k_hist2<<<(N + 255) / 256, 256, 0, stream>>>(keys, N, ctrl, hist2);
    k_find<<<1, 1024, 0, stream>>>(hist2, R, 1, ctrl);

    // 4) stable node compaction
    k_node_count<<<nbN, 256, 0, stream>>>(keys, N, ctrl, nBlkCnt);
    k_scan_blocks<<<1, 1024, 0, stream>>>(nBlkCnt, nBlkOff, nbN, nullptr, nullptr);
    k_node_scatter<<<nbN, 256, 0, stream>>>(keys, N, NK, ctrl, nBlkOff,
                                            mapping, poolIdx, outPool);

    // 5) gathers
    k_gather_x<<<(int)(((long long)NK * 32 + 255) / 256), 256, 0, stream>>>(x, poolIdx, outX, NK);
    k_gather_pos<<<(int)(((long long)NK * 3 + 255) / 256), 256, 0, stream>>>(pos, poolIdx, outPos, NK);

    // 6) stable edge partition + remap + attr permute
    k_edge_count<<<nbE, 256, 0, stream>>>(ei, mapping, E, eBlkCnt);
    k_scan_blocks<<<1, 1024, 0, stream>>>(eBlkCnt, eBlkOff, nbE,
                                          &ctrl[CTRL_NKEPT_E], outCnt);
    k_edge_scatter<<<nbE, 256, 0, stream>>>(ei, mapping, E, eBlkOff, ctrl, dstIdx, outEI);
    k_attr_scatter<<<(int)((E * 4 + 255) / 256), 256, 0, stream>>>(attr, dstIdx, outAttr, E);
}